// Dendrite_755914244697
// MI455X (gfx1250) — compile-verified
//
#include <hip/hip_runtime.h>
#include <math.h>

// Problem constants (from reference setup_inputs)
#define IMG  128   // x is (1, 3, 128, 128)
#define OUTD 124   // 128 - 5 + 1
#define NUM  25
#define SIDE 5
#define CH   3
#define JT   31    // j-positions per block (124 / 4)

// out[c,i,j,n] = log( prod_{a,b} ( atan(10*(x[c,i+a,j+b]*w[...] - q[...]))/pi + 1.1 ) )
__global__ __launch_bounds__(256) void dendrite_kernel(
    const float* __restrict__ x,
    const float* __restrict__ w,
    const float* __restrict__ q,
    float* __restrict__ out)
{
    __shared__ float sx[SIDE * IMG];   // 5 image rows = 2560 B LDS

    const int i   = blockIdx.x;        // output row  0..123
    const int c   = blockIdx.y;        // channel     0..2
    const int j0  = blockIdx.z * JT;   // j tile base
    const int tid = threadIdx.x;

    // ---- Stage x rows [i, i+5) of channel c into LDS via CDNA5 async copy ----
    // 5 rows * 128 floats = 160 x 16B chunks; lanes 0..159 issue one b128 each.
    if (tid < SIDE * (IMG / 4)) {
        const int r = tid >> 5;        // row in 0..4   (32 chunks per row)
        const int m = tid & 31;        // 16B chunk index within the row
        const float* gp = x + ((size_t)c * IMG + (size_t)(i + r)) * IMG + m * 4;
        unsigned lds_addr = (unsigned)(uintptr_t)(&sx[r * IMG + m * 4]);
        asm volatile("global_load_async_to_lds_b128 %0, %1, off"
                     :: "v"(lds_addr), "v"(gp)
                     : "memory");
    }
    asm volatile("s_wait_asynccnt 0x0" ::: "memory");
    __syncthreads();

    const size_t rowbase = ((size_t)c * OUTD + (size_t)i) * OUTD * NUM;
    constexpr float INV_PI = 0.31830988618379067f;

    // Each block covers JT*NUM = 775 outputs; 256 threads stride over them.
    for (int o = tid; o < JT * NUM; o += 256) {
        const int jl = o / NUM;
        const int n  = o - jl * NUM;
        const int j  = j0 + jl;
        const size_t obase = rowbase + (size_t)j * NUM + (size_t)n;
        const float* __restrict__ wp = w + obase * (SIDE * SIDE);
        const float* __restrict__ qp = q + obase * (SIDE * SIDE);

        float prod = 1.0f;
        #pragma unroll
        for (int a = 0; a < SIDE; ++a) {
            #pragma unroll
            for (int b = 0; b < SIDE; ++b) {
                const float xv = sx[a * IMG + j + b];                       // ds_load (hot, reused)
                const float wv = __builtin_nontemporal_load(wp + a * SIDE + b); // streamed, TH=NT
                const float qv = __builtin_nontemporal_load(qp + a * SIDE + b); // streamed, TH=NT
                const float t  = 10.0f * __builtin_fmaf(xv, wv, -qv);
                const float y  = __builtin_fmaf(atanf(t), INV_PI, 1.1f);
                prod *= y;
            }
        }
        out[obase] = logf(prod);   // single log per output (== sum of logs in reference)
    }
}

extern "C" void kernel_launch(void* const* d_in, const int* in_sizes, int n_in,
                              void* d_out, int out_size, void* d_ws, size_t ws_size,
                              hipStream_t stream) {
    (void)in_sizes; (void)n_in; (void)d_ws; (void)ws_size; (void)out_size;
    const float* x = (const float*)d_in[0];
    const float* w = (const float*)d_in[1];
    const float* q = (const float*)d_in[2];
    float* out = (float*)d_out;

    dim3 grid(OUTD, CH, OUTD / JT);   // (124, 3, 4) = 1488 blocks
    dendrite_kernel<<<grid, 256, 0, stream>>>(x, w, q, out);
}